// GCN_Base_25804163514759
// MI455X (gfx1250) — compile-verified
//
#include <hip/hip_runtime.h>

typedef __attribute__((ext_vector_type(2))) float v2f;
typedef __attribute__((ext_vector_type(8))) float v8f;

#define DIM_IN  128
#define DIM_HID 64

// ---------------- degree / normalization ----------------
__global__ void k_deg_init(float* deg, int n) {
    int i = blockIdx.x * blockDim.x + threadIdx.x;
    if (i < n) deg[i] = 1.0f;                       // self-loop
}

__global__ void k_deg_edges(const long long* __restrict__ dst, float* deg, int E) {
    int e = blockIdx.x * blockDim.x + threadIdx.x;
    if (e < E) unsafeAtomicAdd(&deg[(int)dst[e]], 1.0f);
}

__global__ void k_dinv(float* deg, int n) {
    int i = blockIdx.x * blockDim.x + threadIdx.x;
    if (i < n) deg[i] = rsqrtf(deg[i]);             // deg >= 1 always
}

// ---------------- h1 = x @ W1 via V_WMMA_F32_16X16X4_F32 ----------------
// block = 128 threads (4 waves). Each wave computes a 16x64 output strip:
// 4 accumulators (N-tiles of 16), K=128 as 32 steps of K=4.
__global__ __launch_bounds__(128) void k_gemm1(const float* __restrict__ x,
                                               const float* __restrict__ W,
                                               float* __restrict__ h1, int n) {
    __shared__ float sW[DIM_IN * DIM_HID];          // 32 KB, W1 row-major [128][64]
    int tid = threadIdx.x;
    for (int i = tid * 4; i < DIM_IN * DIM_HID; i += 128 * 4)
        *(float4*)&sW[i] = *(const float4*)&W[i];
    __syncthreads();

    int lane = tid & 31;
    int wid  = tid >> 5;
    int rowBase = blockIdx.x * 64 + wid * 16;       // 16 rows per wave
    int m    = lane & 15;                           // row within tile
    int kb   = (lane >> 4) << 1;                    // K pair: 0,1 (lanes 0-15) / 2,3 (16-31)
    int row  = rowBase + m;
    int rowc = row < n ? row : n - 1;               // clamp loads; stores are guarded
    const float* xr = x + (size_t)rowc * DIM_IN + kb;

    v8f acc0 = {}, acc1 = {}, acc2 = {}, acc3 = {};
#pragma unroll
    for (int kk = 0; kk < 32; ++kk) {
        v2f a = *(const v2f*)(xr + kk * 4);         // A: K={kb,kb+1} of row m (8B aligned)
        const float* w0 = &sW[(kk * 4 + kb) * DIM_HID + m];
        const float* w1 = w0 + DIM_HID;
        v2f b0 = { w0[0],  w1[0]  };                // B: N striped across lanes, K in VGPRs
        v2f b1 = { w0[16], w1[16] };
        v2f b2 = { w0[32], w1[32] };
        v2f b3 = { w0[48], w1[48] };
        acc0 = __builtin_amdgcn_wmma_f32_16x16x4_f32(false, a, false, b0, (short)0, acc0, false, false);
        acc1 = __builtin_amdgcn_wmma_f32_16x16x4_f32(false, a, false, b1, (short)0, acc1, false, false);
        acc2 = __builtin_amdgcn_wmma_f32_16x16x4_f32(false, a, false, b2, (short)0, acc2, false, false);
        acc3 = __builtin_amdgcn_wmma_f32_16x16x4_f32(false, a, false, b3, (short)0, acc3, false, false);
    }

    // D layout: VGPR v holds M=v (lanes 0-15) / M=v+8 (lanes 16-31), N=lane%16
    int rq = rowBase + ((lane >> 4) << 3);
#pragma unroll
    for (int v = 0; v < 8; ++v) {
        int r = rq + v;
        if (r < n) {
            float* hr = h1 + (size_t)r * DIM_HID + m;
            hr[0]  = acc0[v];
            hr[16] = acc1[v];
            hr[32] = acc2[v];
            hr[48] = acc3[v];
        }
    }
}

// ---------------- layer-1 aggregation ----------------
__global__ void k_self1(const float* __restrict__ h1, const float* __restrict__ dinv,
                        float* __restrict__ agg, long long total) {
    long long idx = (long long)blockIdx.x * blockDim.x + threadIdx.x;
    if (idx < total) {
        int node = (int)(idx >> 6);
        float di = dinv[node];
        agg[idx] = h1[idx] * di * di;               // self-loop term, also inits agg
    }
}

// one wave per edge: coalesced 64-float row gather + hardware f32 atomics
__global__ __launch_bounds__(256) void k_edge1(const long long* __restrict__ src,
                                               const long long* __restrict__ dst,
                                               const float* __restrict__ h1,
                                               const float* __restrict__ dinv,
                                               float* __restrict__ agg, int E) {
    int lane = threadIdx.x & 31;
    int e = blockIdx.x * (blockDim.x >> 5) + (threadIdx.x >> 5);
    if (e >= E) return;
    int s = (int)src[e], d = (int)dst[e];
    float nrm = dinv[s] * dinv[d];
    const float* hs = h1 + (size_t)s * DIM_HID;
    float* ad = agg + (size_t)d * DIM_HID;
    unsafeAtomicAdd(&ad[lane],      hs[lane]      * nrm);
    unsafeAtomicAdd(&ad[lane + 32], hs[lane + 32] * nrm);
}

// ---------------- t = relu(agg + b1) @ W2 (wave-per-node reduction) ----------------
__global__ __launch_bounds__(256) void k_reduce1(const float* __restrict__ agg,
                                                 const float* __restrict__ b1,
                                                 const float* __restrict__ W2,
                                                 float* __restrict__ t, int n) {
    int lane = threadIdx.x & 31;
    int node = blockIdx.x * (blockDim.x >> 5) + (threadIdx.x >> 5);
    if (node >= n) return;
    const float* a = agg + (size_t)node * DIM_HID;
    float v = fmaxf(a[lane]      + b1[lane],      0.0f) * W2[lane]
            + fmaxf(a[lane + 32] + b1[lane + 32], 0.0f) * W2[lane + 32];
#pragma unroll
    for (int off = 16; off; off >>= 1) v += __shfl_xor(v, off, 32);
    if (lane == 0) t[node] = v;
}

// ---------------- layer-2 aggregation (scalar) ----------------
__global__ void k_self2(const float* __restrict__ t, const float* __restrict__ dinv,
                        const float* __restrict__ b2, float* __restrict__ out, int n) {
    int i = blockIdx.x * blockDim.x + threadIdx.x;
    if (i < n) {
        float di = dinv[i];
        out[i] = b2[0] + t[i] * di * di;            // init + self-loop + bias
    }
}

__global__ void k_edge2(const long long* __restrict__ src, const long long* __restrict__ dst,
                        const float* __restrict__ t, const float* __restrict__ dinv,
                        float* __restrict__ out, int E) {
    int e = blockIdx.x * blockDim.x + threadIdx.x;
    if (e < E) {
        int s = (int)src[e], d = (int)dst[e];
        unsafeAtomicAdd(&out[d], t[s] * dinv[s] * dinv[d]);
    }
}

extern "C" void kernel_launch(void* const* d_in, const int* in_sizes, int n_in,
                              void* d_out, int out_size, void* d_ws, size_t ws_size,
                              hipStream_t stream) {
    const float*     x   = (const float*)d_in[0];
    const long long* e   = (const long long*)d_in[1];   // int64 [2][E]
    const float*     W1  = (const float*)d_in[2];
    const float*     b1  = (const float*)d_in[3];
    const float*     W2  = (const float*)d_in[4];
    const float*     b2  = (const float*)d_in[5];
    float*           out = (float*)d_out;

    int n = in_sizes[0] / DIM_IN;
    int E = in_sizes[1] / 2;
    const long long* src = e;
    const long long* dst = e + E;

    // workspace layout (floats): dinv[n] | h1[n*64] | agg[n*64] | t[n]  (~52 MB)
    float* F    = (float*)d_ws;
    float* dinv = F;
    float* h1   = dinv + n;
    float* agg  = h1 + (size_t)n * DIM_HID;
    float* t    = agg + (size_t)n * DIM_HID;

    k_deg_init <<<(n + 255) / 256, 256, 0, stream>>>(dinv, n);
    k_deg_edges<<<(E + 255) / 256, 256, 0, stream>>>(dst, dinv, E);
    k_dinv     <<<(n + 255) / 256, 256, 0, stream>>>(dinv, n);

    k_gemm1    <<<(n + 63) / 64, 128, 0, stream>>>(x, W1, h1, n);

    long long total = (long long)n * DIM_HID;
    k_self1    <<<(unsigned)((total + 255) / 256), 256, 0, stream>>>(h1, dinv, agg, total);
    k_edge1    <<<(E + 7) / 8, 256, 0, stream>>>(src, dst, h1, dinv, agg, E);

    k_reduce1  <<<(n + 7) / 8, 256, 0, stream>>>(agg, b1, W2, t, n);

    k_self2    <<<(n + 255) / 256, 256, 0, stream>>>(t, dinv, b2, out, n);
    k_edge2    <<<(E + 255) / 256, 256, 0, stream>>>(src, dst, t, dinv, out, E);
}